// GraphModel_15015205666995
// MI455X (gfx1250) — compile-verified
//
#include <hip/hip_runtime.h>
#include <math.h>

// ---------------------------------------------------------------------------
// Problem constants (match reference)
// ---------------------------------------------------------------------------
#define N_NODES 50000
#define N_EDGES 600000
#define DIM     128        // D_IN == D_HID == 128
#define N_IDX   4096
#define ND      (N_NODES * DIM)   // 6,400,000 floats per node-feature buffer

typedef __attribute__((ext_vector_type(2))) float v2f;
typedef __attribute__((ext_vector_type(8))) float v8f;

// ---------------------------------------------------------------------------
// zero a float buffer (float4 stores; n must be a multiple of 4)
// ---------------------------------------------------------------------------
__global__ void zero_f32(float* __restrict__ p, int n4) {
    int t = blockIdx.x * blockDim.x + threadIdx.x;
    if (t < n4) {
        ((float4*)p)[t] = make_float4(0.f, 0.f, 0.f, 0.f);
    }
}

// ---------------------------------------------------------------------------
// degree accumulation: deg[dst] += 1 for every edge of one relation
// ei layout: [2, E] flat -> ei[0..E-1] = src, ei[E..2E-1] = dst
// ---------------------------------------------------------------------------
__global__ void accum_deg(const int* __restrict__ ei, float* __restrict__ deg) {
    int e = blockIdx.x * blockDim.x + threadIdx.x;
    if (e < N_EDGES) {
        atomicAdd(&deg[ei[N_EDGES + e]], 1.0f);
    }
}

// dinv = rsqrt(deg_edges + 1)   (+1 = self loop; deg >= 1 always)
__global__ void finalize_dinv(float* __restrict__ deg, int n) {
    int t = blockIdx.x * blockDim.x + threadIdx.x;
    if (t < n) {
        deg[t] = rsqrtf(deg[t] + 1.0f);
    }
}

// ---------------------------------------------------------------------------
// H = A(MxK=128) @ W(128x128) using V_WMMA_F32_16X16X4_F32.
// Block = 256 threads = 8 waves. Block covers 16 rows x 128 cols; wave w owns
// the 16x16 tile at columns [16w, 16w+16). K=128 -> 32 wmma steps of K=4.
//
// A layout (ISA 7.12.2, 32-bit A 16x4):  lane<16 -> K=0,1 ; lane>=16 -> K=2,3
// for row M = lane%16.  B mirrors with N = lane%16 across lanes.
// C/D layout: VGPR v, lane l -> C[v + 8*(l>>4)][l&15].
// M (=50000) is an exact multiple of 16: EXEC all-1s, as WMMA requires.
// ---------------------------------------------------------------------------
__global__ void __launch_bounds__(256)
gemm_wmma_f32(const float* __restrict__ A, const float* __restrict__ W,
              float* __restrict__ H) {
    const int lane = threadIdx.x & 31;
    const int wave = threadIdx.x >> 5;      // 0..7 -> N tile
    const int m16  = lane & 15;
    const int hi   = lane >> 4;             // 0 or 1
    const int row0 = blockIdx.x * 16;
    const int ncol = (wave << 4) + m16;     // column index for B / C / D

    const float* __restrict__ arow = A + (size_t)(row0 + m16) * DIM;

    v8f c = {};
#pragma unroll
    for (int kk = 0; kk < DIM; kk += 4) {
        const int k0 = kk + 2 * hi;
        v2f a;
        a.x = arow[k0];
        a.y = arow[k0 + 1];
        v2f b;
        b.x = W[(size_t)k0 * DIM + ncol];
        b.y = W[(size_t)(k0 + 1) * DIM + ncol];
        // 8 args: (neg_a, A, neg_b, B, c_mod, C, reuse_a, reuse_b)
        c = __builtin_amdgcn_wmma_f32_16x16x4_f32(
                false, a, false, b, (short)0, c, false, false);
    }

    float* __restrict__ hbase = H + (size_t)row0 * DIM + ncol;
#pragma unroll
    for (int v = 0; v < 8; ++v) {
        const int r = v + 8 * hi;           // row within the 16-row tile
        hbase[(size_t)r * DIM] = c[v];
    }
}

// ---------------------------------------------------------------------------
// Edge scatter: one wave per edge; 32 lanes x float4 = 128 columns.
// agg[dst] += h[src] * dinv[src] * dinv[dst]   via global_atomic_add_f32
// ---------------------------------------------------------------------------
__global__ void __launch_bounds__(256)
scatter_edges(const int* __restrict__ ei, const float* __restrict__ h,
              const float* __restrict__ dinv, float* __restrict__ agg) {
    const int e = blockIdx.x * 8 + (threadIdx.x >> 5);
    if (e >= N_EDGES) return;
    const int lane = threadIdx.x & 31;

    const int src = ei[e];
    const int dst = ei[N_EDGES + e];
    const float w = dinv[src] * dinv[dst];

    const float4 v = *((const float4*)(h + (size_t)src * DIM) + lane);
    float* __restrict__ ad = agg + (size_t)dst * DIM + lane * 4;
    atomicAdd(ad + 0, v.x * w);   // non-returning -> global_atomic_add_f32
    atomicAdd(ad + 1, v.y * w);
    atomicAdd(ad + 2, v.z * w);
    atomicAdd(ad + 3, v.w * w);
}

// ---------------------------------------------------------------------------
// Per-relation finalize: self loop + bias + ReLU, folded running max.
// out = first ? relu(val) : max(out, relu(val))   (valid: relu >= 0)
// ---------------------------------------------------------------------------
__global__ void __launch_bounds__(256)
finalize_rel(const float* __restrict__ agg, const float* __restrict__ h,
             const float* __restrict__ dinv, const float* __restrict__ bias,
             float* __restrict__ out, int first) {
    const int t = blockIdx.x * blockDim.x + threadIdx.x;
    if (t >= ND) return;
    const int v = t >> 7;         // node
    const int c = t & (DIM - 1);  // feature
    const float dv  = dinv[v];
    float val = agg[t] + h[t] * (dv * dv) + bias[c];
    val = fmaxf(val, 0.0f);
    out[t] = first ? val : fmaxf(out[t], val);
}

// ---------------------------------------------------------------------------
// Final gather: d_out[i, :] = src[index[i], :]
// ---------------------------------------------------------------------------
__global__ void gather_out(const float* __restrict__ src,
                           const int* __restrict__ idx,
                           float* __restrict__ out) {
    const int t = blockIdx.x * blockDim.x + threadIdx.x;
    if (t >= N_IDX * DIM) return;
    const int i = t >> 7;
    const int c = t & (DIM - 1);
    out[t] = src[(size_t)idx[i] * DIM + c];
}

// ---------------------------------------------------------------------------
// Orchestration
// ---------------------------------------------------------------------------
extern "C" void kernel_launch(void* const* d_in, const int* in_sizes, int n_in,
                              void* d_out, int out_size, void* d_ws, size_t ws_size,
                              hipStream_t stream) {
    (void)in_sizes; (void)n_in; (void)out_size; (void)ws_size;

    // setup_inputs() dict order:
    // 0:x 1:syn_ei 2:seq_ei 3:sem_ei 4:index
    // then per relation r in {syn,seq,sem}: w1, b1, w2, b2
    const float* x     = (const float*)d_in[0];
    const int*   ei[3] = { (const int*)d_in[1], (const int*)d_in[2], (const int*)d_in[3] };
    const int*   index = (const int*)d_in[4];
    const float* W1[3] = { (const float*)d_in[5],  (const float*)d_in[9],  (const float*)d_in[13] };
    const float* B1[3] = { (const float*)d_in[6],  (const float*)d_in[10], (const float*)d_in[14] };
    const float* W2[3] = { (const float*)d_in[7],  (const float*)d_in[11], (const float*)d_in[15] };
    const float* B2[3] = { (const float*)d_in[8],  (const float*)d_in[12], (const float*)d_in[16] };
    float* out = (float*)d_out;

    // workspace layout (floats): dinv[3*N] | hbuf[ND] | agg[ND] | hcomb[ND] | outcomb[ND]
    float* ws      = (float*)d_ws;
    float* dinv    = ws;                     // 3 * N_NODES (deg -> dinv in place)
    float* hbuf    = dinv + 3 * N_NODES;
    float* agg     = hbuf + ND;
    float* hcomb   = agg + ND;
    float* outcomb = hcomb + ND;

    const int T = 256;
    const dim3 blk(T);
    const int gZeroDinv = (3 * N_NODES / 4 + T - 1) / T;
    const int gDeg      = (N_EDGES + T - 1) / T;
    const int gDinv     = (3 * N_NODES + T - 1) / T;
    const int gGemm     = N_NODES / 16;          // 3125, exact
    const int gZeroAgg  = (ND / 4 + T - 1) / T;
    const int gScat     = (N_EDGES + 7) / 8;     // 8 edges (waves) per block
    const int gFin      = (ND + T - 1) / T;
    const int gGather   = (N_IDX * DIM + T - 1) / T;

    // --- normalization (shared by both layers) ---
    zero_f32<<<gZeroDinv, blk, 0, stream>>>(dinv, 3 * N_NODES / 4);
    for (int r = 0; r < 3; ++r)
        accum_deg<<<gDeg, blk, 0, stream>>>(ei[r], dinv + r * N_NODES);
    finalize_dinv<<<gDinv, blk, 0, stream>>>(dinv, 3 * N_NODES);

    // --- layer 1: x -> hcomb ---
    for (int r = 0; r < 3; ++r) {
        gemm_wmma_f32<<<gGemm, blk, 0, stream>>>(x, W1[r], hbuf);
        zero_f32<<<gZeroAgg, blk, 0, stream>>>(agg, ND / 4);
        scatter_edges<<<gScat, blk, 0, stream>>>(ei[r], hbuf, dinv + r * N_NODES, agg);
        finalize_rel<<<gFin, blk, 0, stream>>>(agg, hbuf, dinv + r * N_NODES, B1[r],
                                               hcomb, r == 0 ? 1 : 0);
    }

    // --- layer 2: hcomb -> outcomb ---
    for (int r = 0; r < 3; ++r) {
        gemm_wmma_f32<<<gGemm, blk, 0, stream>>>(hcomb, W2[r], hbuf);
        zero_f32<<<gZeroAgg, blk, 0, stream>>>(agg, ND / 4);
        scatter_edges<<<gScat, blk, 0, stream>>>(ei[r], hbuf, dinv + r * N_NODES, agg);
        finalize_rel<<<gFin, blk, 0, stream>>>(agg, hbuf, dinv + r * N_NODES, B2[r],
                                               outcomb, r == 0 ? 1 : 0);
    }

    // --- gather selected rows ---
    gather_out<<<gGather, blk, 0, stream>>>(outcomb, index, out);
}